// Mamba3ScanBlock_33706903339416
// MI455X (gfx1250) — compile-verified
//
#include <hip/hip_runtime.h>
#include <hip/hip_bf16.h>
#include <math.h>

// ---------------------------------------------------------------------------
// Mamba3 scan block for MI455X (gfx1250, wave32).
//
// Strategy (see analysis): 5 kernels on `stream`:
//   K1 k_proj    : WMMA f32 16x16x4 GEMMs -> x_branch, z, dt(softplus), B, C
//   K2 k_chunkop : per 256-step chunk, compose per-channel 16x16 transition T_c
//                  and zero-init state b_c (the step operator is linear in h)
//   K3 k_combine : single block folds 256 chunk operators sequentially,
//                  writes per-chunk entry states + h_final (global_prefetch)
//   K4 k_scan    : chunks re-run the elementwise scan from correct entry
//                  state in parallel, emit y_t
//   K5 k_out     : y*silu(z)+D*xb, then out_proj via WMMA f32 16x16x4
// ---------------------------------------------------------------------------

typedef __attribute__((ext_vector_type(2))) float v2f;
typedef __attribute__((ext_vector_type(8))) float v8f;

#define NTOK   65536
#define DM     8
#define DI     16
#define CHUNK  256
#define NCHUNK 256
#define N16    (NTOK * DI)   // 1,048,576 floats per (token, d_inner) array

__device__ inline v8f wmma4(v2f a, v2f b, v8f c) {
  // V_WMMA_F32_16X16X4_F32:  D(16x16 f32) = A(16x4) * B(4x16) + C
  // 8-arg pattern: (neg_a, A, neg_b, B, c_mod, C, reuse_a, reuse_b)
  return __builtin_amdgcn_wmma_f32_16x16x4_f32(
      false, a, false, b, (short)0, c, false, false);
}

__device__ inline float softplus_f(float s) {
  return (s > 20.f) ? s : logf(1.f + __expf(s));
}

// ---------------------------------------------------------------------------
// K1: projections. One wave per 16-token tile (4096 tiles).
// A layout (f32 16x16x4): lane 0-15 -> M=lane, VGPR r holds K = r + 2*(lane>>4)
// B layout (4x16):       lane%16 -> N,      VGPR v holds K = v + 2*(lane>>4)
// C/D layout:            VGPR r, lane l -> (M = r + 8*(l>>4), N = l&15)
// ---------------------------------------------------------------------------
__global__ void k_proj(const float* __restrict__ x,     // (N, 8)
                       const float* __restrict__ inw,   // (32, 8)
                       const float* __restrict__ dtw,   // (16, 16)
                       const float* __restrict__ dtb,   // (16)
                       const float* __restrict__ bw,    // (16, 16)
                       const float* __restrict__ cw,    // (16, 16)
                       float* __restrict__ xb, float* __restrict__ z,
                       float* __restrict__ dt, float* __restrict__ Bv,
                       float* __restrict__ Cv) {
  __shared__ float lds[4][256];
  const int lane = threadIdx.x & 31;
  const int wave = threadIdx.x >> 5;
  const int t0   = (blockIdx.x * 4 + wave) * 16;
  const int lo   = lane & 15, hi = lane >> 4;

  // xz = x @ in_proj_w^T   (K = 8 -> 2 k-steps)
  v8f axb = {}, az = {};
#pragma unroll
  for (int kk = 0; kk < 2; ++kk) {
    const int kb = kk * 4 + 2 * hi;
    v2f a, b0, b1;
    a.x  = x[(t0 + lo) * 8 + kb + 0];
    a.y  = x[(t0 + lo) * 8 + kb + 1];
    b0.x = inw[lo * 8 + kb + 0];          // o = lo        (x_branch half)
    b0.y = inw[lo * 8 + kb + 1];
    b1.x = inw[(lo + 16) * 8 + kb + 0];   // o = lo + 16   (z half)
    b1.y = inw[(lo + 16) * 8 + kb + 1];
    axb = wmma4(a, b0, axb);
    az  = wmma4(a, b1, az);
  }

  // Re-layout x_branch C-tile -> row-major in LDS so it can be reloaded as an
  // A operand (same-wave DS ops are in-order; no barrier needed).
#pragma unroll
  for (int r = 0; r < 8; ++r) lds[wave][(r + 8 * hi) * 16 + lo] = axb[r];

  // dt/B/C projections: x_branch(16x16) @ W^T (K = 16 -> 4 k-steps)
  v8f adt = {}, ab = {}, ac = {};
#pragma unroll
  for (int kk = 0; kk < 4; ++kk) {
    const int kb = kk * 4 + 2 * hi;
    v2f a, b;
    a.x = lds[wave][lo * 16 + kb + 0];
    a.y = lds[wave][lo * 16 + kb + 1];
    b.x = dtw[lo * 16 + kb + 0]; b.y = dtw[lo * 16 + kb + 1];
    adt = wmma4(a, b, adt);
    b.x = bw[lo * 16 + kb + 0];  b.y = bw[lo * 16 + kb + 1];
    ab  = wmma4(a, b, ab);
    b.x = cw[lo * 16 + kb + 0];  b.y = cw[lo * 16 + kb + 1];
    ac  = wmma4(a, b, ac);
  }

  const float bias = dtb[lo];
#pragma unroll
  for (int r = 0; r < 8; ++r) {
    const int idx = (t0 + r + 8 * hi) * 16 + lo;   // (token, i) row-major
    xb[idx] = axb[r];
    z[idx]  = az[r];
    dt[idx] = softplus_f(adt[r] + bias);
    Bv[idx] = ab[r];
    Cv[idx] = ac[r];
  }
}

// ---------------------------------------------------------------------------
// K2: per-chunk transition operator. Block = 16 waves (wave d = inner chan).
// Per channel the step operator on h[d,:] (16-vector) is
//   M[j,j] = abar_j cos_j,  M[j,j-1 mod 16] = -abar_j sin_j
// T (16x16) lives in 8 VGPRs: lane l holds column k=l&15, rows j=r+8*(l>>4).
// Row j-1 for r=0 lives in VGPR7 of the other lane-half -> shfl_xor 16.
// ---------------------------------------------------------------------------
__global__ void k_chunkop(const float* __restrict__ dt,
                          const float* __restrict__ xb,
                          const float* __restrict__ Bv,
                          const float* __restrict__ A_log,
                          const float* __restrict__ rope,
                          float* __restrict__ Tmat,   // (NCHUNK,16,16,16)
                          float* __restrict__ bvec) { // (NCHUNK,16,16)
  const int c    = blockIdx.x;
  const int d    = threadIdx.x >> 5;
  const int lane = threadIdx.x & 31;
  const int lo   = lane & 15, hi = lane >> 4;

  float Arow[8], Frow[8];
#pragma unroll
  for (int r = 0; r < 8; ++r) {
    const int j = r + 8 * hi;
    Arow[r] = -__expf(A_log[d * 16 + j]);
    Frow[r] = rope[d * 16 + j];
  }
  const float A2 = -__expf(A_log[d * 16 + lo]);
  const float F2 = rope[d * 16 + lo];

  float T[8];
#pragma unroll
  for (int r = 0; r < 8; ++r) T[r] = ((r + 8 * hi) == lo) ? 1.f : 0.f;
  float b = 0.f;

  const int t0  = c * CHUNK;
  const int src = (lane & 16) | ((lo + 15) & 15);   // j-1 (mod 16) in-half

  for (int t = 0; t < CHUNK; ++t) {
    const int tt   = t0 + t;
    const float dtc = dt[tt * 16 + d];
    const float xbd = xb[tt * 16 + d];
    const float Bj  = Bv[tt * 16 + lo];

    // T <- M_t * T   (update r=7..1 with old T[r-1]; r=0 uses wrapped row 15/7)
    const float roll0 = __shfl_xor(T[7], 16, 32);
#pragma unroll
    for (int r = 7; r >= 1; --r) {
      const float ta   = 0.5f * dtc * Arow[r];
      const float abar = (1.f + ta) / (1.f - ta + 1e-8f);
      const float ang  = dtc * Frow[r];
      T[r] = abar * (__cosf(ang) * T[r] - __sinf(ang) * T[r - 1]);
    }
    {
      const float ta   = 0.5f * dtc * Arow[0];
      const float abar = (1.f + ta) / (1.f - ta + 1e-8f);
      const float ang  = dtc * Frow[0];
      T[0] = abar * (__cosf(ang) * T[0] - __sinf(ang) * roll0);
    }
    // b <- M_t b + u_t   (lane holds b[j=lo], duplicated across halves)
    const float ta2   = 0.5f * dtc * A2;
    const float abar2 = (1.f + ta2) / (1.f - ta2 + 1e-8f);
    const float ang2  = dtc * F2;
    const float brl   = __shfl(b, src, 32);
    b = abar2 * (__cosf(ang2) * b - __sinf(ang2) * brl) + dtc * Bj * xbd;
  }

#pragma unroll
  for (int r = 0; r < 8; ++r)
    Tmat[((c * 16 + d) * 16 + (r + 8 * hi)) * 16 + lo] = T[r];
  if (hi == 0) bvec[(c * 16 + d) * 16 + lo] = b;
}

// ---------------------------------------------------------------------------
// K3: sequential fold over 256 chunks. 1 block, 16 waves (wave d = channel).
// Lane j computes h_new[j] = b[j] + sum_k T[j,k] h[k]; h[k] via shfl.
// ---------------------------------------------------------------------------
__global__ void k_combine(const float* __restrict__ Tmat,
                          const float* __restrict__ bvec,
                          const float* __restrict__ init_h,
                          float* __restrict__ hstart,     // (NCHUNK,16,16)
                          float* __restrict__ h_final) {  // d_out + N*DM
  const int d    = threadIdx.x >> 5;
  const int lane = threadIdx.x & 31;
  const int lo   = lane & 15;
  float h = init_h[d * 16 + lo];

  for (int c = 0; c < NCHUNK; ++c) {
    if (c + 1 < NCHUNK) {   // hide the serial-load latency of the next chunk
      __builtin_prefetch(Tmat + (((c + 1) * 16 + d) * 16 + lo) * 16, 0, 1);
      __builtin_prefetch(bvec + ((c + 1) * 16 + d) * 16 + lo, 0, 1);
    }
    if (lane < 16) hstart[(c * 16 + d) * 16 + lo] = h;   // entry state

    const float4* Trow =
        reinterpret_cast<const float4*>(Tmat + ((c * 16 + d) * 16 + lo) * 16);
    const float4 q0 = Trow[0], q1 = Trow[1], q2 = Trow[2], q3 = Trow[3];
    float Tr[16] = {q0.x, q0.y, q0.z, q0.w, q1.x, q1.y, q1.z, q1.w,
                    q2.x, q2.y, q2.z, q2.w, q3.x, q3.y, q3.z, q3.w};
    float hn = bvec[(c * 16 + d) * 16 + lo];
#pragma unroll
    for (int k = 0; k < 16; ++k)
      hn = fmaf(Tr[k], __shfl(h, (lane & 16) | k, 32), hn);
    h = hn;
  }
  if (lane < 16) h_final[d * 16 + lo] = h;
}

// ---------------------------------------------------------------------------
// K4: parallel re-scan per chunk from the correct entry state, emitting y.
// ---------------------------------------------------------------------------
__global__ void k_scan(const float* __restrict__ dt,
                       const float* __restrict__ xb,
                       const float* __restrict__ Bv,
                       const float* __restrict__ Cv,
                       const float* __restrict__ A_log,
                       const float* __restrict__ rope,
                       const float* __restrict__ hstart,
                       float* __restrict__ y) {
  const int c    = blockIdx.x;
  const int d    = threadIdx.x >> 5;
  const int lane = threadIdx.x & 31;
  const int lo   = lane & 15;
  const float A2 = -__expf(A_log[d * 16 + lo]);
  const float F2 = rope[d * 16 + lo];
  float h = hstart[(c * 16 + d) * 16 + lo];
  const int t0  = c * CHUNK;
  const int src = (lane & 16) | ((lo + 15) & 15);

  for (int t = 0; t < CHUNK; ++t) {
    const int tt    = t0 + t;
    const float dtc = dt[tt * 16 + d];
    const float xbd = xb[tt * 16 + d];
    const float Bj  = Bv[tt * 16 + lo];
    const float Cj  = Cv[tt * 16 + lo];
    const float ta   = 0.5f * dtc * A2;
    const float abar = (1.f + ta) / (1.f - ta + 1e-8f);
    const float ang  = dtc * F2;
    const float hr   = __shfl(h, src, 32);
    h = abar * (__cosf(ang) * h - __sinf(ang) * hr) + dtc * Bj * xbd;
    float p = h * Cj;                         // y[d] = sum_j h[d,j] C[j]
    p += __shfl_xor(p, 1, 32);
    p += __shfl_xor(p, 2, 32);
    p += __shfl_xor(p, 4, 32);
    p += __shfl_xor(p, 8, 32);
    if (lane == 0) y[tt * 16 + d] = p;
  }
}

// ---------------------------------------------------------------------------
// K5: gated epilogue + out_proj (16 -> 8, N padded to 16) via WMMA f32.
// ---------------------------------------------------------------------------
__global__ void k_out(const float* __restrict__ y,
                      const float* __restrict__ z,
                      const float* __restrict__ xb,
                      const float* __restrict__ Dp,
                      const float* __restrict__ outw,   // (8,16)
                      float* __restrict__ out) {        // (N,8)
  const int lane = threadIdx.x & 31;
  const int wave = threadIdx.x >> 5;
  const int t0   = (blockIdx.x * 4 + wave) * 16;
  const int lo   = lane & 15, hi = lane >> 4;

  v8f acc = {};
#pragma unroll
  for (int kk = 0; kk < 4; ++kk) {
    const int kb = kk * 4 + 2 * hi;
    v2f a, b;
#pragma unroll
    for (int v = 0; v < 2; ++v) {
      const int k   = kb + v;
      const int idx = (t0 + lo) * 16 + k;
      const float zz  = z[idx];
      const float sil = zz / (1.f + __expf(-zz));
      const float val = y[idx] * sil + Dp[k] * xb[idx];
      if (v == 0) a.x = val; else a.y = val;
    }
    b.x = (lo < 8) ? outw[lo * 16 + kb + 0] : 0.f;   // pad N to 16 with zeros
    b.y = (lo < 8) ? outw[lo * 16 + kb + 1] : 0.f;
    acc = wmma4(a, b, acc);
  }
  if (lo < 8) {
#pragma unroll
    for (int r = 0; r < 8; ++r) out[(t0 + r + 8 * hi) * 8 + lo] = acc[r];
  }
}

// ---------------------------------------------------------------------------
extern "C" void kernel_launch(void* const* d_in, const int* in_sizes, int n_in,
                              void* d_out, int out_size, void* d_ws,
                              size_t ws_size, hipStream_t stream) {
  const float* x    = (const float*)d_in[0];
  const float* init = (const float*)d_in[1];
  const float* inw  = (const float*)d_in[2];
  const float* dtw  = (const float*)d_in[3];
  const float* dtb  = (const float*)d_in[4];
  const float* bw   = (const float*)d_in[5];
  const float* cw   = (const float*)d_in[6];
  const float* alog = (const float*)d_in[7];
  const float* Dp   = (const float*)d_in[8];
  const float* rope = (const float*)d_in[9];
  const float* outw = (const float*)d_in[10];
  float* out = (float*)d_out;

  // Workspace layout (floats). Total ~28.5 MB (ws_size assumed >= that).
  float* ws = (float*)d_ws;
  float* XB = ws;              // N*16
  float* Z  = XB + N16;        // N*16
  float* DT = Z + N16;         // N*16
  float* BV = DT + N16;        // N*16
  float* CV = BV + N16;        // N*16
  float* Y  = CV + N16;        // N*16
  float* T  = Y + N16;         // NCHUNK*16*16*16 = N16
  float* BC = T + N16;         // NCHUNK*16*16 = 65536
  float* HS = BC + NCHUNK * 256;  // NCHUNK*16*16 = 65536

  k_proj<<<NTOK / 64, 128, 0, stream>>>(x, inw, dtw, dtb, bw, cw,
                                        XB, Z, DT, BV, CV);
  k_chunkop<<<NCHUNK, 512, 0, stream>>>(DT, XB, BV, alog, rope, T, BC);
  k_combine<<<1, 512, 0, stream>>>(T, BC, init, HS, out + NTOK * DM);
  k_scan<<<NCHUNK, 512, 0, stream>>>(DT, XB, BV, CV, alog, rope, HS, Y);
  k_out<<<NTOK / 64, 128, 0, stream>>>(Y, Z, XB, Dp, outw, out);
}